// CausalDepthwiseConv1d_15169824489976
// MI455X (gfx1250) — compile-verified
//
#include <hip/hip_runtime.h>

// ---------------------------------------------------------------------------
// Causal depthwise Conv1d, x:[B=8, L=4096, C=1024] f32, K=7, channels-last.
// Memory-bound (AI ~1.75 F/B, ~268 MB floor => ~11.5 us at 23.3 TB/s).
// Strategy: async global->LDS DMA double buffering (CDNA5 ASYNCcnt path,
// global_load_async_to_lds_b128) + float4/wave32 coalesced streaming +
// register rolling window along L + nontemporal output stores.
// WMMA is inapplicable: depthwise conv's summand x[l+k-6,c]*w[c,k] cannot be
// factored as A[m,k]*B[k,n] (x depends jointly on l,k AND c).
// 512 blocks x 2-per-WGP (128KB LDS each) to cover the full chip.
// ---------------------------------------------------------------------------

typedef float v4f __attribute__((ext_vector_type(4)));
typedef int   v4i __attribute__((ext_vector_type(4)));

#ifndef __has_builtin
#define __has_builtin(x) 0
#endif

#if __has_builtin(__builtin_amdgcn_global_load_async_to_lds_b128)
#define HAVE_ASYNC 1
// Signature (verified via hipcc diagnostic): (AS1 v4i* src, AS3 v4i* dst, Ii, Ii)
#define ASYNC_COPY_B128(gsrc, ldst)                                          \
  __builtin_amdgcn_global_load_async_to_lds_b128(                            \
      (__attribute__((address_space(1))) v4i*)(void*)(gsrc),                 \
      (__attribute__((address_space(3))) v4i*)(void*)(ldst), 0, 0)
#else
#define HAVE_ASYNC 0
// Fallback: synchronous copy through VGPRs (still correct; no ASYNCcnt).
#define ASYNC_COPY_B128(gsrc, ldst)                                          \
  do { *(v4f*)(ldst) = *(const v4f*)(gsrc); } while (0)
#endif

#if __has_builtin(__builtin_amdgcn_s_wait_asynccnt)
#define WAIT_ASYNC_LE(n) __builtin_amdgcn_s_wait_asynccnt(n)
#elif HAVE_ASYNC
#define WAIT_ASYNC_LE(n) asm volatile("s_wait_asynccnt %0" ::"i"(n) : "memory")
#else
#define WAIT_ASYNC_LE(n) do {} while (0)
#endif

#if __has_builtin(__builtin_amdgcn_s_wait_dscnt)
#define WAIT_DS0() __builtin_amdgcn_s_wait_dscnt(0)
#else
#define WAIT_DS0() asm volatile("s_wait_dscnt 0" ::: "memory")
#endif

// Problem constants (fixed by the reference harness).
constexpr int Bc   = 8;
constexpr int Lc   = 4096;
constexpr int Cc   = 1024;   // floats per row; 256 threads * float4
constexpr int Kc   = 7;
// Tiling: each block owns one 64-row segment of L, split into 4 tiles of 16
// rows, double buffered in LDS (2 * 16 * 4KB = 128KB -> 2 workgroups/WGP).
// 8 batches * 64 segments = 512 blocks -> ~256 WGPs * 2 blocks.
// Halo re-reads (6/64 = 9.4%) hit L2 (192MB holds all of x), not HBM.
constexpr int SPAN = 64;
constexpr int TL   = 16;
constexpr int NT   = SPAN / TL;   // 4 tiles per block
constexpr int SEGS = Lc / SPAN;   // 64 segments per batch

__global__ __launch_bounds__(256) void
causal_dwconv1d_kernel(const float* __restrict__ x,
                       const float* __restrict__ w,
                       const float* __restrict__ bias,
                       float* __restrict__ out)
{
  __shared__ float lds[2 * TL * Cc];   // 131072 bytes

  const int tid = threadIdx.x;         // 0..255
  const int c4  = tid * 4;             // this thread's 4 channels
  const int b   = blockIdx.x / SEGS;
  const int l0  = (blockIdx.x % SEGS) * SPAN;

  // Per-thread per-channel tap weights: wt[k] = {w[c4+j][k]}_{j=0..3}
  v4f wt[Kc];
  #pragma unroll
  for (int k = 0; k < Kc; ++k) {
    #pragma unroll
    for (int j = 0; j < 4; ++j) {
      wt[k][j] = w[(c4 + j) * Kc + k];
    }
  }
  const v4f bias4 = *(const v4f*)(bias + c4);

  const float* xb = x   + (size_t)b * Lc * Cc;
  float*       ob = out + (size_t)b * Lc * Cc;

  // Rolling window: rows l-6..l-1 relative to the next output row.
  v4f win[6];
  if (l0 == 0) {
    v4f z; z.x = z.y = z.z = z.w = 0.0f;
    #pragma unroll
    for (int i = 0; i < 6; ++i) win[i] = z;
  } else {
    #pragma unroll
    for (int i = 0; i < 6; ++i)
      win[i] = *(const v4f*)(xb + (size_t)(l0 - 6 + i) * Cc + c4);
  }

  // Issue a tile's 16 async b128 loads: rows [l0+t*TL, l0+t*TL+TL) of this
  // thread's 4-channel column slice into LDS buffer bi.
  auto issue_tile = [&](int t, int bi) {
    const float* src = xb + (size_t)(l0 + t * TL) * Cc + c4;
    float*       dst = &lds[bi * TL * Cc + c4];
    #pragma unroll
    for (int r = 0; r < TL; ++r)
      ASYNC_COPY_B128(src + (size_t)r * Cc, dst + r * Cc);
  };

  // Consume one tile: 1 ds_load_b128 + 28 FMA + 1 nontemporal b128 store per
  // row, rotating the 7-tap window through registers.
  auto compute_tile = [&](int t, int bi) {
    const float* lsrc = &lds[bi * TL * Cc + c4];
    float*       dst  = ob + (size_t)(l0 + t * TL) * Cc + c4;
    #pragma unroll
    for (int r = 0; r < TL; ++r) {
      v4f cur = *(const v4f*)(lsrc + r * Cc);   // row l (tap k=6)
      v4f acc = bias4;
      #pragma unroll
      for (int k = 0; k < 6; ++k) acc += win[k] * wt[k];
      acc += cur * wt[6];
      __builtin_nontemporal_store(acc, (v4f*)(dst + (size_t)r * Cc));
      #pragma unroll
      for (int k = 0; k < 5; ++k) win[k] = win[k + 1];
      win[5] = cur;
    }
  };

  // Software pipeline: two tiles in flight on the async engine at all times.
  // ASYNCcnt completes in order, so "wait <= TL" proves the older tile landed.
  issue_tile(0, 0);
  issue_tile(1, 1);
  #pragma unroll
  for (int t = 0; t < NT; ++t) {
    if (t < NT - 1) {
      WAIT_ASYNC_LE(TL);   // tile t landed; tile t+1 still in flight
    } else {
      WAIT_ASYNC_LE(0);    // last tile: drain fully
    }
    compute_tile(t, t & 1);
    if (t + 2 < NT) {
      WAIT_DS0();          // all ds reads of this buffer retired before reuse
      issue_tile(t + 2, t & 1);
    }
  }
}

extern "C" void kernel_launch(void* const* d_in, const int* in_sizes, int n_in,
                              void* d_out, int out_size, void* d_ws, size_t ws_size,
                              hipStream_t stream) {
  (void)in_sizes; (void)n_in; (void)d_ws; (void)ws_size; (void)out_size;
  const float* x    = (const float*)d_in[0];
  const float* w    = (const float*)d_in[1];
  const float* bias = (const float*)d_in[2];
  float*       out  = (float*)d_out;

  dim3 grid(Bc * SEGS);   // 512 blocks
  dim3 block(256);        // 8 waves (wave32), 4 channels/thread covers C=1024
  hipLaunchKernelGGL(causal_dwconv1d_kernel, grid, block, 0, stream,
                     x, w, bias, out);
}